// MultiHeadedAttention_4604204941604
// MI455X (gfx1250) — compile-verified
//
#include <hip/hip_runtime.h>
#include <hip/hip_bf16.h>

// ---------------------------------------------------------------------------
// MultiHeadedAttention with distance-MLP reweighting, CDNA5 (gfx1250) WMMA.
// B=2, N=1024, D_MODEL=512, H=8, D_K=64, HID=50 (padded to 64).
// All math fp32 via v_wmma_f32_16x16x4_f32 (wave32, 16x16 C tiles).
// Attention K/V/Q tiles staged to LDS via GLOBAL_LOAD_ASYNC_TO_LDS_B128
// (ASYNCcnt / s_wait_asynccnt), double-buffered across the kt sweep.
// ---------------------------------------------------------------------------

#define BB   2
#define NN_  1024
#define DM   512
#define NH   8
#define DK   64
#define NEGV (-1.0e8f)

typedef float v2f __attribute__((ext_vector_type(2)));
typedef float v8f __attribute__((ext_vector_type(8)));

#if __has_builtin(__builtin_amdgcn_global_load_async_to_lds_b128)
#define ASYNC_LDS 1
// Builtin signature (probe: hipcc diagnostic): first arg is a pointer to a
// 16-byte int vector in the global (__device__/as1) address space; second is
// the LDS (as3) destination; then imm offset and imm cpol.
typedef int v4i_ __attribute__((vector_size(16)));
typedef __attribute__((address_space(1))) v4i_* as1_v4i;
typedef __attribute__((address_space(3))) v4i_* as3_v4i;
#else
#define ASYNC_LDS 0
#endif

__device__ __forceinline__ void wait_async0() {
#if ASYNC_LDS
#if __has_builtin(__builtin_amdgcn_s_wait_asynccnt)
  __builtin_amdgcn_s_wait_asynccnt(0);
#else
  asm volatile("s_wait_asynccnt 0x0" ::: "memory");
#endif
#endif
}

// Copy 16 bytes (4 floats) global -> LDS, async on gfx1250 if available.
__device__ __forceinline__ void async_copy16(const float* g, float* l) {
#if ASYNC_LDS
  __builtin_amdgcn_global_load_async_to_lds_b128(
      (as1_v4i)(void*)g, (as3_v4i)(void*)l, /*offset=*/0, /*cpol=*/0);
#else
  #pragma unroll
  for (int u = 0; u < 4; ++u) l[u] = g[u];
#endif
}

__device__ __forceinline__ v8f zero8() {
  v8f z = {0.f, 0.f, 0.f, 0.f, 0.f, 0.f, 0.f, 0.f};
  return z;
}

__device__ __forceinline__ v8f wmma_f32(v2f a, v2f b, v8f c) {
  // D = A(16x4) * B(4x16) + C(16x16), fp32, wave32.
  return __builtin_amdgcn_wmma_f32_16x16x4_f32(
      /*neg_a=*/false, a, /*neg_b=*/false, b,
      /*c_mod=*/(short)0, c, /*reuse_a=*/false, /*reuse_b=*/false);
}

// ---------------------------------------------------------------------------
// Generic fp32 WMMA GEMM tile: each wave computes a 32x32 block of
// C[M,N] = X[M,K] @ W[K,N]; 8 waves -> 128x64 block tile.
// interleave=1 scatters output to [B,H,N,DK] head-split layout.
// ---------------------------------------------------------------------------
__device__ __forceinline__ void store_frag(float* Out, v8f c, int rbase, int cbase,
                                           int lane, int Ndim, int interleave) {
  const int M0 = (lane >> 4) * 8;
  const int n  = cbase + (lane & 15);
  #pragma unroll
  for (int r = 0; r < 8; ++r) {
    const int row = rbase + M0 + r;
    const float v = c[r];
    if (interleave) {
      const int b  = row >> 10;          // row = b*N + nn
      const int nn = row & (NN_ - 1);
      const int hh = n >> 6;             // n = hh*DK + dd
      const int dd = n & (DK - 1);
      Out[(((size_t)(b * NH + hh)) * NN_ + nn) * DK + dd] = v;
    } else {
      Out[(size_t)row * Ndim + n] = v;
    }
  }
}

__device__ __forceinline__ void gemm_tile(const float* X, const float* W, float* Out,
                                          int Kdim, int Ndim, int interleave) {
  const int t    = threadIdx.x;
  const int lane = t & 31;
  const int wave = t >> 5;
  const int wy   = wave >> 1;
  const int wx   = wave & 1;
  const int rm   = blockIdx.x * 128 + wy * 32;
  const int cn   = blockIdx.y * 64 + wx * 32;
  const int m    = lane & 15;
  const int kh   = (lane >> 4) << 1;      // 0 for lanes 0-15, 2 for lanes 16-31

  v8f c00 = zero8(), c01 = zero8(), c10 = zero8(), c11 = zero8();

  for (int k0 = 0; k0 < Kdim; k0 += 4) {
    const int kb = k0 + kh;
    const float* x0 = X + (size_t)(rm + m) * Kdim + kb;
    const float* x1 = x0 + (size_t)16 * Kdim;
    const float* w0 = W + (size_t)kb * Ndim + cn + m;
    const float* w1 = w0 + Ndim;
    if ((k0 & 63) == 0 && k0 + 64 < Kdim) {
      __builtin_prefetch(x0 + 64, 0, 1);                  // global_prefetch_b8
      __builtin_prefetch(w0 + (size_t)64 * Ndim, 0, 1);
    }
    v2f a0 = { x0[0], x0[1] };
    v2f a1 = { x1[0], x1[1] };
    v2f b0 = { w0[0],  w1[0]  };
    v2f b1 = { w0[16], w1[16] };
    c00 = wmma_f32(a0, b0, c00);
    c01 = wmma_f32(a0, b1, c01);
    c10 = wmma_f32(a1, b0, c10);
    c11 = wmma_f32(a1, b1, c11);
  }
  store_frag(Out, c00, rm,      cn,      lane, Ndim, interleave);
  store_frag(Out, c01, rm,      cn + 16, lane, Ndim, interleave);
  store_frag(Out, c10, rm + 16, cn,      lane, Ndim, interleave);
  store_frag(Out, c11, rm + 16, cn + 16, lane, Ndim, interleave);
}

// grid (16, 8, 3): z selects Q/K/V; outputs head-split [B,H,N,DK].
__global__ __launch_bounds__(256) void qkv_proj_kernel(
    const float* __restrict__ q, const float* __restrict__ k, const float* __restrict__ v,
    const float* __restrict__ Wq, const float* __restrict__ Wk, const float* __restrict__ Wv,
    float* __restrict__ Qo, float* __restrict__ Ko, float* __restrict__ Vo) {
  const float* X; const float* W; float* O;
  if (blockIdx.z == 0)      { X = q; W = Wq; O = Qo; }
  else if (blockIdx.z == 1) { X = k; W = Wk; O = Ko; }
  else                      { X = v; W = Wv; O = Vo; }
  gemm_tile(X, W, O, DM, DM, /*interleave=*/1);
}

// grid (16, 8): out = ATTN[2048,512] @ Wo[512,512]
__global__ __launch_bounds__(256) void out_proj_kernel(
    const float* __restrict__ X, const float* __restrict__ Wo, float* __restrict__ O) {
  gemm_tile(X, Wo, O, DM, DM, /*interleave=*/0);
}

// ---------------------------------------------------------------------------
// Distance MLP: per pair (b,i,j): relu(d*W1+b1) -> relu(@W2+b2) -> relu(@W3+b3)
// -> @W4+b4, masked, squared. Hidden padded 50 -> 64; layers 2/3 are WMMA.
// Each wave owns 16 rows; block = 8 waves = 128 rows. LDS: one 64x64 weight
// buffer (reloaded per layer) + one 16x64 activation buffer per wave.
// ---------------------------------------------------------------------------
__device__ __forceinline__ void mlp_layer_wmma(float* h /*in/out 16x64*/,
                                               const float* sW /*64x64*/,
                                               const float* sb /*64*/, int lane) {
  const int m  = lane & 15;
  const int kh = (lane >> 4) << 1;
  v8f c0 = zero8(), c1 = zero8(), c2 = zero8(), c3 = zero8();
  for (int k0 = 0; k0 < 64; k0 += 4) {
    const int kb = k0 + kh;
    v2f a  = { h[m * 64 + kb], h[m * 64 + kb + 1] };
    const float* w0 = sW + kb * 64 + m;
    const float* w1 = w0 + 64;
    v2f b0 = { w0[0],  w1[0]  };
    v2f b1 = { w0[16], w1[16] };
    v2f b2 = { w0[32], w1[32] };
    v2f b3 = { w0[48], w1[48] };
    c0 = wmma_f32(a, b0, c0);
    c1 = wmma_f32(a, b1, c1);
    c2 = wmma_f32(a, b2, c2);
    c3 = wmma_f32(a, b3, c3);
  }
  // All LDS reads feeding the WMMAs are issued before any store below; a wave
  // is a single instruction stream, so the in-place update is safe.
  const int M0 = (lane >> 4) * 8;
  const int n  = lane & 15;
  #pragma unroll
  for (int r = 0; r < 8; ++r) {
    const int row = M0 + r;
    h[row * 64 + n]      = fmaxf(c0[r] + sb[n],      0.f);
    h[row * 64 + 16 + n] = fmaxf(c1[r] + sb[16 + n], 0.f);
    h[row * 64 + 32 + n] = fmaxf(c2[r] + sb[32 + n], 0.f);
    h[row * 64 + 48 + n] = fmaxf(c3[r] + sb[48 + n], 0.f);
  }
}

__global__ __launch_bounds__(256) void dist_mlp_kernel(
    const float* __restrict__ dsrc, const int* __restrict__ mask,
    const float* __restrict__ dW1, const float* __restrict__ db1,
    const float* __restrict__ dW2, const float* __restrict__ db2,
    const float* __restrict__ dW3, const float* __restrict__ db3,
    const float* __restrict__ dW4, const float* __restrict__ db4,
    float* __restrict__ dsq) {
  __shared__ float sW[64 * 64];                 // reloaded: W2 then W3
  __shared__ float sW1[64], sb1[64], sb2[64], sb3[64], sW4[64];
  __shared__ float sH[8][16 * 64];              // per-wave activations

  const int t    = threadIdx.x;
  const int lane = t & 31;
  const int wave = t >> 5;

  if (t < 64) {
    sW1[t] = (t < 50) ? dW1[t] : 0.f;
    sb1[t] = (t < 50) ? db1[t] : 0.f;
    sb2[t] = (t < 50) ? db2[t] : 0.f;
    sb3[t] = (t < 50) ? db3[t] : 0.f;
    sW4[t] = (t < 50) ? dW4[t] : 0.f;
  }
  for (int i = t; i < 64 * 64; i += 256) {
    const int kk = i >> 6, nn = i & 63;
    sW[i] = (kk < 50 && nn < 50) ? dW2[kk * 50 + nn] : 0.f;
  }
  __syncthreads();

  const size_t row0 = (size_t)blockIdx.x * 128 + (size_t)wave * 16;
  float* h = sH[wave];

  // Layer 1: h[m][j] = relu(d[m]*W1[j] + b1[j]) (cols >=50 land at 0).
  for (int i = lane; i < 16 * 64; i += 32) {
    const int mrow = i >> 6, j = i & 63;
    const float d = dsrc[row0 + mrow];
    h[i] = fmaxf(d * sW1[j] + sb1[j], 0.f);
  }
  __syncthreads();

  // Layer 2 (WMMA, in-place).
  mlp_layer_wmma(h, sW, sb2, lane);
  __syncthreads();

  // Swap weights to W3.
  for (int i = t; i < 64 * 64; i += 256) {
    const int kk = i >> 6, nn = i & 63;
    sW[i] = (kk < 50 && nn < 50) ? dW3[kk * 50 + nn] : 0.f;
  }
  __syncthreads();

  // Layer 3 (WMMA, in-place).
  mlp_layer_wmma(h, sW, sb3, lane);
  __syncthreads();

  // Layer 4 + masking + square. Lanes 0..15 each finish one row.
  if (lane < 16) {
    float acc = db4[0];
    for (int j = 0; j < 64; ++j) acc += h[lane * 64 + j] * sW4[j];
    const size_t p = row0 + lane;
    const int b   = (int)(p >> 20);             // N*N = 2^20
    const int rem = (int)(p & ((1u << 20) - 1));
    const int i   = rem >> 10;
    const int j   = rem & (NN_ - 1);
    const bool ok = (mask[b * NN_ + i] != 0) && (mask[b * NN_ + j] != 0);
    const float dv = ok ? acc : 0.f;
    dsq[p] = dv * dv;
  }
}

// ---------------------------------------------------------------------------
// Flash-style attention. One wave = one 16-query tile of one (b,h); the 8
// waves of a block share (b,h) and sweep the same key tiles, so K/V tiles are
// staged once per block in LDS via async-to-LDS, double-buffered. S = Q@K^T
// via WMMA (scale folded into scores); diagonal/mask -> NEG; online softmax;
// P *= dsq; P staged to per-wave LDS (C->A layout); O += P@V via WMMA.
// ---------------------------------------------------------------------------
__global__ __launch_bounds__(256) void attn_kernel(
    const float* __restrict__ Q, const float* __restrict__ K, const float* __restrict__ V,
    const float* __restrict__ dsq, const int* __restrict__ mask,
    float* __restrict__ attnbuf) {
  __shared__ float sQ[8][16 * 64];   // per-wave Q tile
  __shared__ float sP[8][16 * 16];   // per-wave P staging
  __shared__ float sK[2][16 * 64];   // block-shared, double-buffered
  __shared__ float sV[2][16 * 64];

  const int t    = threadIdx.x;
  const int lane = t & 31;
  const int wave = t >> 5;
  const int bh   = blockIdx.x;          // 0..15
  const int b    = bh >> 3;
  const int h    = bh & 7;
  const int qt   = blockIdx.y * 8 + wave;
  const int q0   = qt * 16;

  const float* Qh = Q + (size_t)bh * NN_ * DK;
  const float* Kh = K + (size_t)bh * NN_ * DK;
  const float* Vh = V + (size_t)bh * NN_ * DK;
  const float scale = 1.0f / (8.0f + 1e-8f); // 1/(sqrt(dk)+1e-8)

  float* q = sQ[wave];
  float* P = sP[wave];

  // Issue async stage of K/V tile 0 (one b128 per thread per matrix).
  {
    const int t4 = t * 4;                       // 256 thr * 16B = 4KB tile
    async_copy16(Kh + t4, &sK[0][t4]);
    async_copy16(Vh + t4, &sV[0][t4]);
  }
  // Async stage of this wave's Q tile (contiguous 4KB; unscaled).
  for (int i = lane * 4; i < 16 * 64; i += 128)
    async_copy16(Qh + (size_t)q0 * DK + i, &q[i]);

  // 16-bit query validity mask, replicated to all lanes.
  unsigned qmb;
  {
    const int qi = q0 + (lane & 15);
    const unsigned long long bal = __ballot(mask[b * NN_ + qi] != 0);
    qmb = (unsigned)bal & 0xFFFFu;
  }
  wait_async0();
  __syncthreads();

  float mrow[8], lrow[8];
  #pragma unroll
  for (int r = 0; r < 8; ++r) { mrow[r] = -3.0e38f; lrow[r] = 0.f; }
  v8f o0 = zero8(), o1 = zero8(), o2 = zero8(), o3 = zero8();

  const int mm = lane & 15;
  const int kh = (lane >> 4) << 1;
  const int M0 = (lane >> 4) * 8;
  const float* drow = dsq + ((size_t)b << 20);

  for (int kt = 0; kt < NN_ / 16; ++kt) {
    const int k0g = kt * 16;
    const int cur = kt & 1;

    // Prefetch next K/V tile into the other buffer (overlaps compute).
    if (kt + 1 < NN_ / 16) {
      const int t4 = t * 4;
      const size_t base = (size_t)(k0g + 16) * DK;
      async_copy16(Kh + base + t4, &sK[cur ^ 1][t4]);
      async_copy16(Vh + base + t4, &sV[cur ^ 1][t4]);
    }
    const float* sKc = sK[cur];
    const float* sVc = sV[cur];

    // ---- S tile = Q @ K^T (16x16), K = 64 -> 16 WMMA steps (LDS operands).
    v8f s = zero8();
    for (int k0 = 0; k0 < 64; k0 += 4) {
      const int kb = k0 + kh;
      v2f a  = { q[mm * 64 + kb],   q[mm * 64 + kb + 1]   };
      v2f bf = { sKc[mm * 64 + kb], sKc[mm * 64 + kb + 1] };
      s = wmma_f32(a, bf, s);
    }

    // ---- scale + masks + dsq gather (column = lane&15, rows M0..M0+7).
    const int kglob = k0g + mm;
    const int kmv   = mask[b * NN_ + kglob];
    float dsqv[8];
    #pragma unroll
    for (int r = 0; r < 8; ++r) {
      const int row   = M0 + r;
      const int qglob = q0 + row;
      const bool valid = (kmv != 0) && ((qmb >> row) & 1u);
      float sv = s[r] * scale;
      if (!valid || kglob == qglob) sv = NEGV;
      s[r] = sv;
      dsqv[r] = drow[(size_t)qglob * NN_ + kglob];
    }

    // ---- online softmax per row (reduce over 16-lane group).
    float pv[8];
    #pragma unroll
    for (int r = 0; r < 8; ++r) {
      float vmax = s[r];
      #pragma unroll
      for (int off = 1; off < 16; off <<= 1)
        vmax = fmaxf(vmax, __shfl_xor(vmax, off));
      const float mnew = fmaxf(mrow[r], vmax);
      const float sc   = __expf(mrow[r] - mnew);
      const float p    = __expf(s[r] - mnew);
      float ps = p;
      #pragma unroll
      for (int off = 1; off < 16; off <<= 1)
        ps += __shfl_xor(ps, off);
      lrow[r] = lrow[r] * sc + ps;
      mrow[r] = mnew;
      o0[r] *= sc; o1[r] *= sc; o2[r] *= sc; o3[r] *= sc;
      pv[r] = p * dsqv[r];   // post-softmax reweight (denominator unaffected)
    }

    // ---- stage P (C-layout -> LDS) then consume as A-layout for P @ V.
    #pragma unroll
    for (int r = 0; r < 8; ++r) P[(M0 + r) * 16 + mm] = pv[r];
    __builtin_amdgcn_wave_barrier();   // keep LDS write->read order (per-wave)

    for (int k0 = 0; k0 < 16; k0 += 4) {
      const int kb = k0 + kh;
      v2f a = { P[mm * 16 + kb], P[mm * 16 + kb + 1] };
      const float* v0 = sVc + kb * 64;
      const float* v1 = v0 + 64;
      v2f b0 = { v0[mm],      v1[mm]      };
      v2f b1 = { v0[16 + mm], v1[16 + mm] };
      v2f b2 = { v0[32 + mm], v1[32 + mm] };
      v2f b3 = { v0[48 + mm], v1[48 + mm] };
      o0 = wmma_f32(a, b0, o0);
      o1 = wmma_f32(a, b1, o1);
      o2 = wmma_f32(a, b2, o2);
      o3 = wmma_f32(a, b3, o3);
    }

    // Drain this iteration's async prefetch, then rotate buffers.
    wait_async0();
    __syncthreads();
  }

  // ---- finalize: normalize, query-mask, write concat-head layout [B,N,H*DK].
  #pragma unroll
  for (int r = 0; r < 8; ++r) {
    const int row   = M0 + r;
    const int qglob = q0 + row;
    const float inv = ((qmb >> row) & 1u) ? (1.0f / lrow[r]) : 0.0f;
    float* dst = attnbuf + ((size_t)b * NN_ + qglob) * DM + h * DK;
    dst[mm]      = o0[r] * inv;
    dst[16 + mm] = o1[r] * inv;
    dst[32 + mm] = o2[r] * inv;
    dst[48 + mm] = o3[r] * inv;
  }
}

// ---------------------------------------------------------------------------
// Host launcher.
// ---------------------------------------------------------------------------
extern "C" void kernel_launch(void* const* d_in, const int* in_sizes, int n_in,
                              void* d_out, int out_size, void* d_ws, size_t ws_size,
                              hipStream_t stream) {
  (void)in_sizes; (void)n_in; (void)out_size; (void)ws_size;

  const float* query = (const float*)d_in[0];
  const float* key_  = (const float*)d_in[1];
  const float* value = (const float*)d_in[2];
  const float* dists = (const float*)d_in[3];
  const int*   mask  = (const int*)  d_in[4];
  const float* Wq    = (const float*)d_in[5];
  const float* Wk    = (const float*)d_in[6];
  const float* Wv    = (const float*)d_in[7];
  const float* Wo    = (const float*)d_in[8];
  const float* dW1   = (const float*)d_in[9];
  const float* db1   = (const float*)d_in[10];
  const float* dW2   = (const float*)d_in[11];
  const float* db2   = (const float*)d_in[12];
  const float* dW3   = (const float*)d_in[13];
  const float* db3   = (const float*)d_in[14];
  const float* dW4   = (const float*)d_in[15];
  const float* db4   = (const float*)d_in[16];

  float* ws = (float*)d_ws;
  const size_t qkv_elems  = (size_t)BB * NH * NN_ * DK;   // 1,048,576
  const size_t dsq_elems  = (size_t)BB * NN_ * NN_;       // 2,097,152
  float* Qb   = ws;
  float* Kb   = Qb + qkv_elems;
  float* Vb   = Kb + qkv_elems;
  float* DSQ  = Vb + qkv_elems;
  float* ATTN = DSQ + dsq_elems;                          // 1,048,576

  // 1) Q/K/V projections -> head-split [B,H,N,DK].
  qkv_proj_kernel<<<dim3(16, 8, 3), 256, 0, stream>>>(
      query, key_, value, Wq, Wk, Wv, Qb, Kb, Vb);

  // 2) distance MLP -> masked squared distances dsq[B,N,N].
  dist_mlp_kernel<<<dim3((BB * NN_ * NN_) / 128), 256, 0, stream>>>(
      dists, mask, dW1, db1, dW2, db2, dW3, db3, dW4, db4, DSQ);

  // 3) attention with dsq reweighting -> ATTN[B,N,H*DK].
  attn_kernel<<<dim3(16, 8), 256, 0, stream>>>(Qb, Kb, Vb, DSQ, mask, ATTN);

  // 4) output projection -> d_out[B,N,DM].
  out_proj_kernel<<<dim3(16, 8), 256, 0, stream>>>(ATTN, Wo, (float*)d_out);
}